// BinarizeLayer_72224170049693
// MI455X (gfx1250) — compile-verified
//
#include <hip/hip_runtime.h>
#include <stdint.h>

// out[b,f] = (medians[f] > 0 && x[b,f] >= medians[f]) ? 1.0f : 0.0f
// Pure HBM-bandwidth problem: ~537 MB traffic -> ~23 us floor @ 23.3 TB/s.
// Strategy: b128 streaming with non-temporal hints, medians staged once per
// block via the CDNA5 async global->LDS path, predicate pre-folded to a single
// compare, and 4-row unrolling for deep per-wave memory-level parallelism.

typedef float v4f __attribute__((ext_vector_type(4)));

#define BLOCK 256
#define VEC 4
#define TILE_W (BLOCK * VEC)  // 1024 floats (4 KiB) of medians per block tile
#define UNROLL 4

__global__ __launch_bounds__(BLOCK)
void binarize_kernel(const float* __restrict__ x,
                     const float* __restrict__ med,
                     float* __restrict__ out,
                     int rows, int cols) {
    __shared__ float smed[TILE_W];

    const int tid  = threadIdx.x;
    const int col0 = blockIdx.x * TILE_W;

    // --- Stage this block's 4 KiB median tile into LDS via the CDNA5 async
    //     global->LDS engine (ASYNCcnt-tracked), b128 per lane.
    {
        uint32_t lds_off = (uint32_t)(uintptr_t)(&smed[0]) + (uint32_t)tid * 16u;
        const float* gsrc = med + col0 + tid * VEC;
        asm volatile(
            "global_load_async_to_lds_b128 %0, %1, off\n\t"
            "s_wait_asynccnt 0"
            :
            : "v"(lds_off), "v"(gsrc)
            : "memory");
    }
    __syncthreads();

    // m' = (m > 0) ? m : +inf collapses the two-term predicate into a single
    // (x >= m') compare (NaN x compares false, matching jnp semantics).
    v4f m = *reinterpret_cast<const v4f*>(&smed[tid * VEC]);
    const float INF = __builtin_inff();
    v4f mp;
    mp.x = (m.x > 0.0f) ? m.x : INF;
    mp.y = (m.y > 0.0f) ? m.y : INF;
    mp.z = (m.z > 0.0f) ? m.z : INF;
    mp.w = (m.w > 0.0f) ? m.w : INF;

    const size_t rowStride  = (size_t)cols;
    const size_t stepStride = rowStride * gridDim.y;   // elems between my rows

    const v4f* xin = reinterpret_cast<const v4f*>(
                         x + (size_t)blockIdx.y * rowStride + col0) + tid;
    v4f* po        = reinterpret_cast<v4f*>(
                         out + (size_t)blockIdx.y * rowStride + col0) + tid;
    const size_t stepV4 = stepStride / VEC;            // step in v4f units

    int r = blockIdx.y;
    const int rstep = gridDim.y;

    // --- Main loop: 4 rows per macro-iteration. 4 b128 loads issued
    //     back-to-back (VMEM clause, 4 outstanding per wave), then 16
    //     cmp/sel (VOPD dual-issue), then 4 b128 stores. Non-temporal so the
    //     512 MB stream doesn't evict hot lines.
    for (; r + (UNROLL - 1) * rstep < rows; r += UNROLL * rstep) {
        v4f v0 = __builtin_nontemporal_load(xin);
        v4f v1 = __builtin_nontemporal_load(xin + stepV4);
        v4f v2 = __builtin_nontemporal_load(xin + 2 * stepV4);
        v4f v3 = __builtin_nontemporal_load(xin + 3 * stepV4);

        v4f o0, o1, o2, o3;
        o0.x = (v0.x >= mp.x) ? 1.0f : 0.0f;
        o0.y = (v0.y >= mp.y) ? 1.0f : 0.0f;
        o0.z = (v0.z >= mp.z) ? 1.0f : 0.0f;
        o0.w = (v0.w >= mp.w) ? 1.0f : 0.0f;
        o1.x = (v1.x >= mp.x) ? 1.0f : 0.0f;
        o1.y = (v1.y >= mp.y) ? 1.0f : 0.0f;
        o1.z = (v1.z >= mp.z) ? 1.0f : 0.0f;
        o1.w = (v1.w >= mp.w) ? 1.0f : 0.0f;
        o2.x = (v2.x >= mp.x) ? 1.0f : 0.0f;
        o2.y = (v2.y >= mp.y) ? 1.0f : 0.0f;
        o2.z = (v2.z >= mp.z) ? 1.0f : 0.0f;
        o2.w = (v2.w >= mp.w) ? 1.0f : 0.0f;
        o3.x = (v3.x >= mp.x) ? 1.0f : 0.0f;
        o3.y = (v3.y >= mp.y) ? 1.0f : 0.0f;
        o3.z = (v3.z >= mp.z) ? 1.0f : 0.0f;
        o3.w = (v3.w >= mp.w) ? 1.0f : 0.0f;

        __builtin_nontemporal_store(o0, po);
        __builtin_nontemporal_store(o1, po + stepV4);
        __builtin_nontemporal_store(o2, po + 2 * stepV4);
        __builtin_nontemporal_store(o3, po + 3 * stepV4);

        xin += UNROLL * stepV4;
        po  += UNROLL * stepV4;
    }

    // --- Tail (not taken for 16384 x 4096 with grid.y == 256).
    for (; r < rows; r += rstep) {
        v4f v = __builtin_nontemporal_load(xin);
        v4f o;
        o.x = (v.x >= mp.x) ? 1.0f : 0.0f;
        o.y = (v.y >= mp.y) ? 1.0f : 0.0f;
        o.z = (v.z >= mp.z) ? 1.0f : 0.0f;
        o.w = (v.w >= mp.w) ? 1.0f : 0.0f;
        __builtin_nontemporal_store(o, po);
        xin += stepV4;
        po  += stepV4;
    }
}

extern "C" void kernel_launch(void* const* d_in, const int* in_sizes, int n_in,
                              void* d_out, int out_size, void* d_ws, size_t ws_size,
                              hipStream_t stream) {
    const float* x   = (const float*)d_in[0];
    const float* med = (const float*)d_in[1];
    float* out       = (float*)d_out;

    const int cols = in_sizes[1];            // 4096
    const int rows = in_sizes[0] / cols;     // 16384

    dim3 block(BLOCK);
    dim3 grid(cols / TILE_W, 256);           // 4 column tiles x 256 row groups
    binarize_kernel<<<grid, block, 0, stream>>>(x, med, out, rows, cols);
}